// PDP_42408507081350
// MI455X (gfx1250) — compile-verified
//
#include <hip/hip_runtime.h>
#include <math.h>

// ---------------------------------------------------------------------------
// ws layout (uint32 words). Single reusable 2048-word histogram keeps the
// scratch footprint at ~8.2 KB regardless of ws_size.
// ---------------------------------------------------------------------------
#define HIST      0
#define C_PREFIX1 2048   // selected 11-bit coarse bucket (bits 30..20 of |w|)
#define C_CA1     2049   // count of elements strictly above that bucket
#define C_PREFIX2 2050   // selected 22-bit prefix (bits 30..9)
#define C_CA2     2051   // count strictly above that prefix
#define C_MAXBEL  2052   // max |w| bit pattern strictly below prefix22
#define C_T2      2053   // t*t as float bits
#define WS_WORDS  2054

__device__ __forceinline__ unsigned absbits(float x) {
    return __float_as_uint(x) & 0x7FFFFFFFu;
}

__global__ __launch_bounds__(256) void k_zero(unsigned* __restrict__ ws) {
    int i = blockIdx.x * blockDim.x + threadIdx.x;
    if (i < WS_WORDS) ws[i] = 0u;
}

// ---------------------------------------------------------------------------
// Pass 1: 2048-bucket histogram of bits[30:20] of |w| (LDS-privatized).
// ---------------------------------------------------------------------------
__global__ __launch_bounds__(256) void k_hist1(const float* __restrict__ w,
                                               unsigned* __restrict__ ws,
                                               long long n4, long long n) {
    __shared__ unsigned h[2048];
    for (int i = threadIdx.x; i < 2048; i += blockDim.x) h[i] = 0u;
    __syncthreads();
    const float4* __restrict__ w4 = (const float4*)w;
    long long stride = (long long)gridDim.x * blockDim.x;
    for (long long i = (long long)blockIdx.x * blockDim.x + threadIdx.x; i < n4; i += stride) {
        float4 v = w4[i];
        atomicAdd(&h[absbits(v.x) >> 20], 1u);
        atomicAdd(&h[absbits(v.y) >> 20], 1u);
        atomicAdd(&h[absbits(v.z) >> 20], 1u);
        atomicAdd(&h[absbits(v.w) >> 20], 1u);
    }
    if (blockIdx.x == 0 && threadIdx.x == 0)
        for (long long i = n4 * 4; i < n; ++i) atomicAdd(&h[absbits(w[i]) >> 20], 1u);
    __syncthreads();
    for (int i = threadIdx.x; i < 2048; i += blockDim.x)
        if (h[i]) atomicAdd(&ws[HIST + i], h[i]);
}

// Scan 1: pick coarse bucket containing descending rank `lim`; re-zero hist.
__global__ void k_scan1(unsigned* __restrict__ ws, long long lim) {
    unsigned long long cum = 0;
    int sel = 0;
    for (int b = 2047; b >= 0; --b) {
        unsigned c = ws[HIST + b];
        if (cum + c > (unsigned long long)lim) { sel = b; break; }
        cum += c;
    }
    ws[C_PREFIX1] = (unsigned)sel;
    ws[C_CA1]     = (unsigned)cum;
    for (int i = 0; i < 2048; ++i) ws[HIST + i] = 0u;
}

// ---------------------------------------------------------------------------
// Pass 2: within the selected coarse bucket, histogram bits[19:9].
// ---------------------------------------------------------------------------
__global__ __launch_bounds__(256) void k_hist2(const float* __restrict__ w,
                                               unsigned* __restrict__ ws,
                                               long long n4, long long n) {
    __shared__ unsigned h[2048];
    for (int i = threadIdx.x; i < 2048; i += blockDim.x) h[i] = 0u;
    __syncthreads();
    const unsigned p1 = ws[C_PREFIX1];
    const float4* __restrict__ w4 = (const float4*)w;
    long long stride = (long long)gridDim.x * blockDim.x;
    for (long long i = (long long)blockIdx.x * blockDim.x + threadIdx.x; i < n4; i += stride) {
        float4 v = w4[i];
        unsigned u;
        u = absbits(v.x); if ((u >> 20) == p1) atomicAdd(&h[(u >> 9) & 2047u], 1u);
        u = absbits(v.y); if ((u >> 20) == p1) atomicAdd(&h[(u >> 9) & 2047u], 1u);
        u = absbits(v.z); if ((u >> 20) == p1) atomicAdd(&h[(u >> 9) & 2047u], 1u);
        u = absbits(v.w); if ((u >> 20) == p1) atomicAdd(&h[(u >> 9) & 2047u], 1u);
    }
    if (blockIdx.x == 0 && threadIdx.x == 0)
        for (long long i = n4 * 4; i < n; ++i) {
            unsigned u = absbits(w[i]);
            if ((u >> 20) == p1) atomicAdd(&h[(u >> 9) & 2047u], 1u);
        }
    __syncthreads();
    for (int i = threadIdx.x; i < 2048; i += blockDim.x)
        if (h[i]) atomicAdd(&ws[HIST + i], h[i]);
}

__global__ void k_scan2(unsigned* __restrict__ ws, long long lim) {
    unsigned long long cum = ws[C_CA1];
    unsigned p1 = ws[C_PREFIX1];
    int sel = 0;
    for (int b = 2047; b >= 0; --b) {
        unsigned c = ws[HIST + b];
        if (cum + c > (unsigned long long)lim) { sel = b; break; }
        cum += c;
    }
    ws[C_PREFIX2] = (p1 << 11) | (unsigned)sel;
    ws[C_CA2]     = (unsigned)cum;
    for (int i = 0; i < 2048; ++i) ws[HIST + i] = 0u;
}

// ---------------------------------------------------------------------------
// Pass 3: within the 22-bit prefix, histogram bits[8:0] (full bit pattern now
// resolved). Also track max pattern strictly below the prefix (for the case
// where rank lim+1 falls outside the prefix).
// ---------------------------------------------------------------------------
__global__ __launch_bounds__(256) void k_hist3(const float* __restrict__ w,
                                               unsigned* __restrict__ ws,
                                               long long n4, long long n) {
    __shared__ unsigned h[512];
    __shared__ unsigned mx;
    for (int i = threadIdx.x; i < 512; i += blockDim.x) h[i] = 0u;
    if (threadIdx.x == 0) mx = 0u;
    __syncthreads();
    const unsigned p2 = ws[C_PREFIX2];
    const unsigned lo = p2 << 9;
    const float4* __restrict__ w4 = (const float4*)w;
    long long stride = (long long)gridDim.x * blockDim.x;
    for (long long i = (long long)blockIdx.x * blockDim.x + threadIdx.x; i < n4; i += stride) {
        float4 v = w4[i];
        unsigned u;
        u = absbits(v.x); if ((u >> 9) == p2) atomicAdd(&h[u & 511u], 1u); else if (u < lo) atomicMax(&mx, u);
        u = absbits(v.y); if ((u >> 9) == p2) atomicAdd(&h[u & 511u], 1u); else if (u < lo) atomicMax(&mx, u);
        u = absbits(v.z); if ((u >> 9) == p2) atomicAdd(&h[u & 511u], 1u); else if (u < lo) atomicMax(&mx, u);
        u = absbits(v.w); if ((u >> 9) == p2) atomicAdd(&h[u & 511u], 1u); else if (u < lo) atomicMax(&mx, u);
    }
    if (blockIdx.x == 0 && threadIdx.x == 0)
        for (long long i = n4 * 4; i < n; ++i) {
            unsigned u = absbits(w[i]);
            if ((u >> 9) == p2) atomicAdd(&h[u & 511u], 1u); else if (u < lo) atomicMax(&mx, u);
        }
    __syncthreads();
    for (int i = threadIdx.x; i < 512; i += blockDim.x)
        if (h[i]) atomicAdd(&ws[HIST + i], h[i]);
    if (threadIdx.x == 0) atomicMax(&ws[C_MAXBEL], mx);
}

// Scan 3: resolve exact Wh (rank lim) and Wt (rank lim+1), emit t^2.
__global__ void k_scan3(unsigned* __restrict__ ws, long long lim) {
    unsigned long long cum = ws[C_CA2];
    unsigned p2 = ws[C_PREFIX2];
    int sel = 0; unsigned selc = 0; unsigned long long selcum = 0;
    for (int b = 511; b >= 0; --b) {
        unsigned c = ws[HIST + b];
        if (cum + c > (unsigned long long)lim) { sel = b; selc = c; selcum = cum; break; }
        cum += c;
    }
    unsigned whPat = (p2 << 9) | (unsigned)sel;
    unsigned wtPat = whPat;
    if (!(selcum + selc > (unsigned long long)(lim + 1))) {
        wtPat = ws[C_MAXBEL];  // fallback: largest pattern below this prefix
        for (int b = sel - 1; b >= 0; --b)
            if (ws[HIST + b]) { wtPat = (p2 << 9) | (unsigned)b; break; }
    }
    float wh = __uint_as_float(whPat);
    float wt = __uint_as_float(wtPat);
    float t  = 0.5f * (wh + wt);
    ws[C_T2] = __float_as_uint(t * t);
}

// ---------------------------------------------------------------------------
// Final pass: out = w / (1 + exp((t^2 - w^2)/1e-5)).
// Streams tiles through LDS with CDNA5 async global->LDS loads. Each wave
// issues 8x b128 per lane into its own LDS slots, waits on ASYNCcnt, and
// consumes its own data — no cross-wave barrier needed.
// ---------------------------------------------------------------------------
__global__ __launch_bounds__(256) void k_final(const float* __restrict__ w,
                                               float* __restrict__ out,
                                               const unsigned* __restrict__ ws,
                                               long long n4, long long n) {
    __shared__ float4 buf[2048];           // 32 KB: 8 waves * 32 lanes * 8 float4
    const float t2 = __uint_as_float(ws[C_T2]);
    const int wave  = threadIdx.x >> 5;
    const int lane  = threadIdx.x & 31;
    const int slot0 = (wave * 32 + lane) * 8;   // this lane's 8 contiguous float4
    const float4* __restrict__ w4 = (const float4*)w;
    float4* __restrict__ o4 = (float4*)out;

    const long long blockTile = 2048;      // float4 per block per iteration
    const long long gridTile  = (long long)gridDim.x * blockTile;

    for (long long base = (long long)blockIdx.x * blockTile; base < n4; base += gridTile) {
#pragma unroll
        for (int k = 0; k < 8; ++k) {
            long long idx = base + (long long)(wave * 256 + k * 32 + lane);
            if (idx < n4) {
                unsigned ldsOff = (unsigned)(unsigned long long)(&buf[slot0 + k]);
                const float4* g = w4 + idx;
                asm volatile("global_load_async_to_lds_b128 %0, %1, off"
                             :: "v"(ldsOff), "v"(g) : "memory");
            }
        }
        asm volatile("s_wait_asynccnt 0x0" ::: "memory");
#pragma unroll
        for (int k = 0; k < 8; ++k) {
            long long idx = base + (long long)(wave * 256 + k * 32 + lane);
            if (idx < n4) {
                float4 v = buf[slot0 + k];
                float4 r;
                r.x = v.x / (1.0f + expf((t2 - v.x * v.x) * 1e5f));
                r.y = v.y / (1.0f + expf((t2 - v.y * v.y) * 1e5f));
                r.z = v.z / (1.0f + expf((t2 - v.z * v.z) * 1e5f));
                r.w = v.w / (1.0f + expf((t2 - v.w * v.w) * 1e5f));
                o4[idx] = r;
            }
        }
    }
    if (blockIdx.x == 0 && threadIdx.x == 0) {
        for (long long i = n4 * 4; i < n; ++i) {
            float v = w[i];
            out[i] = v / (1.0f + expf((t2 - v * v) * 1e5f));
        }
    }
}

extern "C" void kernel_launch(void* const* d_in, const int* in_sizes, int n_in,
                              void* d_out, int out_size, void* d_ws, size_t ws_size,
                              hipStream_t stream) {
    (void)n_in; (void)out_size; (void)ws_size;
    const float* w  = (const float*)d_in[0];
    float* out      = (float*)d_out;
    unsigned* ws    = (unsigned*)d_ws;
    long long n  = (long long)in_sizes[0];
    long long n4 = n >> 2;

    // Replicate Python: lim = max(0, int((1.0 - 0.8) * n) - 1)  (double math)
    long long lim = (long long)((1.0 - 0.8) * (double)n) - 1;
    if (lim < 0) lim = 0;

    k_zero <<<(WS_WORDS + 255) / 256, 256, 0, stream>>>(ws);
    k_hist1<<<2048, 256, 0, stream>>>(w, ws, n4, n);
    k_scan1<<<1, 1, 0, stream>>>(ws, lim);
    k_hist2<<<2048, 256, 0, stream>>>(w, ws, n4, n);
    k_scan2<<<1, 1, 0, stream>>>(ws, lim);
    k_hist3<<<2048, 256, 0, stream>>>(w, ws, n4, n);
    k_scan3<<<1, 1, 0, stream>>>(ws, lim);
    k_final<<<2048, 256, 0, stream>>>(w, out, ws, n4, n);
}